// AttentionMT_65060164600276
// MI455X (gfx1250) — compile-verified
//
#include <hip/hip_runtime.h>
#include <hip/hip_bf16.h>
#include <math.h>

// Problem constants (reference: N=64, S=2048, D=768, fp32)
#define NB     64
#define SLEN   2048
#define DDIM   768
#define CHUNKS 8
#define ROWS_PER_BLOCK (SLEN / CHUNKS)     // 256 rows of one batch per workgroup
#define TILE_ROWS 16
#define TILES  (ROWS_PER_BLOCK / TILE_ROWS) // 16 tiles
#define BLOCK  256
#define DP     (DDIM + 4)                   // padded LDS row stride (772): 16-way bank spread
#define PART_STRIDE (DDIM + 2)              // per-partial: [m, l, acc[768]]

typedef float v2f __attribute__((ext_vector_type(2)));
typedef float v8f __attribute__((ext_vector_type(8)));

__device__ __forceinline__ float sigmoidf_(float x) { return 1.0f / (1.0f + __expf(-x)); }

// ---------------------------------------------------------------------------
// Kernel 1: single-pass online-softmax partial over a 256-row S-chunk.
// grid = (NB, CHUNKS), block = 256 threads (8 wave32).
// ---------------------------------------------------------------------------
__global__ void __launch_bounds__(BLOCK)
attn_flash_partial(const float* __restrict__ hidden,
                   const float* __restrict__ w_att,
                   float* __restrict__ ws)
{
    __shared__ float tile[TILE_ROWS * DP];   // 16 x 772 f32 = 48.25 KB
    __shared__ float watt_s[DDIM];           // 3 KB
    __shared__ float scores_s[TILE_ROWS];

    const int n    = blockIdx.x;
    const int c    = blockIdx.y;
    const int tid  = threadIdx.x;
    const int lane = tid & 31;
    const int wave = tid >> 5;

    // stage w_att once (read each WMMA iteration as the broadcast B fragment)
    for (int i = tid; i < DDIM; i += BLOCK) watt_s[i] = w_att[i];

    const float* hbase =
        hidden + ((size_t)n * SLEN + (size_t)c * ROWS_PER_BLOCK) * DDIM;

    float runM = -INFINITY, runL = 0.0f;
    float acc0 = 0.0f, acc1 = 0.0f, acc2 = 0.0f;   // d = tid, tid+256, tid+512

    for (int t = 0; t < TILES; ++t) {
        const float* src = hbase + (size_t)t * TILE_ROWS * DDIM;

        __syncthreads();   // previous tile's consumers done before overwrite

        // ---- coalesced stage: 3072 float4, 12 per thread, padded row stride
        #pragma unroll
        for (int j = 0; j < (TILE_ROWS * DDIM) / (BLOCK * 4); ++j) {
            int i4  = tid + j * BLOCK;           // float4 index
            int r   = i4 / (DDIM / 4);
            int c4  = i4 % (DDIM / 4);
            float4 v = ((const float4*)src)[i4];
            *(float4*)&tile[r * DP + c4 * 4] = v;
        }
        if (tid < TILE_ROWS) scores_s[tid] = 0.0f;

        // prefetch the next tile while we compute on this one
        if (t + 1 < TILES)
            __builtin_prefetch(src + (size_t)TILE_ROWS * DDIM + tid * 16, 0, 1);

        __syncthreads();

        // ---- scores for the 16 staged rows: s[m] = sum_k tile[m][k]*w_att[k]
        const int m  = lane & 15;            // A row owned by this lane
        const int kb = (lane >> 4) << 1;     // K sub-offset: 0 (lanes 0-15) / 2 (16-31)
        const int kw = wave * (DDIM / 8);    // 8 waves split K=768 -> 96 each

#if __has_builtin(__builtin_amdgcn_wmma_f32_16x16x4_f32)
        v8f cacc = {};
        #pragma unroll 4
        for (int kk = 0; kk < DDIM / 8; kk += 4) {
            const int k0 = kw + kk;
            v2f a, b;
            a.x = tile[m * DP + k0 + kb];        // A[m][k0+kb]
            a.y = tile[m * DP + k0 + kb + 1];    // A[m][k0+kb+1]
            b.x = watt_s[k0 + kb];               // B[k][j] = w_att[k] for all j
            b.y = watt_s[k0 + kb + 1];
            cacc = __builtin_amdgcn_wmma_f32_16x16x4_f32(
                false, a, false, b, (short)0, cacc, false, false);
        }
        // C layout: lane L holds column L%16; VGPR v holds row v + 8*(L/16).
        // All columns are identical (B columns identical) -> lanes 0 & 16 own
        // the 16 scores. Sum the 8 per-wave K-partials with LDS float atomics.
        if (lane == 0) {
            #pragma unroll
            for (int v = 0; v < 8; ++v) atomicAdd(&scores_s[v], cacc[v]);
        }
        if (lane == 16) {
            #pragma unroll
            for (int v = 0; v < 8; ++v) atomicAdd(&scores_s[8 + v], cacc[v]);
        }
#else
        // VALU fallback (keeps the file compiling if the builtin is absent)
        #pragma unroll
        for (int i = 0; i < TILE_ROWS; ++i) {
            float s = 0.0f;
            #pragma unroll
            for (int j = 0; j < 3; ++j) {
                int k = kw + lane + j * 32;
                s = fmaf(tile[i * DP + k], watt_s[k], s);
            }
            atomicAdd(&scores_s[i], s);
        }
#endif
        __syncthreads();

        // ---- online softmax update (block-uniform scalars, per-thread copy)
        float tmax = scores_s[0];
        #pragma unroll
        for (int i = 1; i < TILE_ROWS; ++i) tmax = fmaxf(tmax, scores_s[i]);
        const float newM = fmaxf(runM, tmax);
        const float corr = __expf(runM - newM);   // exp(-inf)=0 on first tile

        float w[TILE_ROWS];
        float lsum = 0.0f;
        #pragma unroll
        for (int i = 0; i < TILE_ROWS; ++i) {
            w[i] = __expf(scores_s[i] - newM);
            lsum += w[i];
        }
        runL = runL * corr + lsum;
        runM = newM;

        acc0 *= corr; acc1 *= corr; acc2 *= corr;
        #pragma unroll
        for (int i = 0; i < TILE_ROWS; ++i) {
            acc0 = fmaf(w[i], tile[i * DP + tid],        acc0);
            acc1 = fmaf(w[i], tile[i * DP + tid + 256],  acc1);
            acc2 = fmaf(w[i], tile[i * DP + tid + 512],  acc2);
        }
    }

    // ---- emit partial: [m, l, acc[768]]
    float* p = ws + ((size_t)n * CHUNKS + c) * PART_STRIDE;
    if (tid == 0) { p[0] = runM; p[1] = runL; }
    p += 2;
    p[tid]       = acc0;
    p[tid + 256] = acc1;
    p[tid + 512] = acc2;
}

// ---------------------------------------------------------------------------
// Kernel 2: combine the 8 chunk-partials per batch, write cvec + the two
// sigmoid heads. grid = NB, block = 256.
// Output layout (tuple order, flat): y_pred[64] | context_pred[64] | cvec[64*768]
// ---------------------------------------------------------------------------
__global__ void __launch_bounds__(BLOCK)
attn_combine(const float* __restrict__ ws,
             const float* __restrict__ ctx,
             const float* __restrict__ w_rec,
             const float* __restrict__ b_rec,
             float* __restrict__ out)
{
    __shared__ float redA[BLOCK];
    __shared__ float redB[BLOCK];

    const int n   = blockIdx.x;
    const int tid = threadIdx.x;
    const float* base = ws + (size_t)n * CHUNKS * PART_STRIDE;

    float M = -INFINITY;
    #pragma unroll
    for (int c = 0; c < CHUNKS; ++c)
        M = fmaxf(M, base[(size_t)c * PART_STRIDE]);

    float L = 0.0f;
    float sc[CHUNKS];
    #pragma unroll
    for (int c = 0; c < CHUNKS; ++c) {
        const float mc = base[(size_t)c * PART_STRIDE];
        sc[c] = __expf(mc - M);
        L += base[(size_t)c * PART_STRIDE + 1] * sc[c];
    }
    const float invL = 1.0f / L;

    float dotR = 0.0f, dotC = 0.0f;
    #pragma unroll
    for (int j = 0; j < 3; ++j) {
        const int d = tid + j * BLOCK;
        float a = 0.0f;
        #pragma unroll
        for (int c = 0; c < CHUNKS; ++c)
            a = fmaf(sc[c], base[(size_t)c * PART_STRIDE + 2 + d], a);
        const float cv = a * invL;
        out[2 * NB + (size_t)n * DDIM + d] = cv;             // cvec
        dotR = fmaf(cv, w_rec[d], dotR);
        dotC = fmaf(cv, ctx[(size_t)n * DDIM + d], dotC);
    }

    redA[tid] = dotR;
    redB[tid] = dotC;
    __syncthreads();
    for (int s = BLOCK / 2; s > 0; s >>= 1) {
        if (tid < s) {
            redA[tid] += redA[tid + s];
            redB[tid] += redB[tid + s];
        }
        __syncthreads();
    }
    if (tid == 0) {
        out[n]      = sigmoidf_(redA[0] + b_rec[0]);   // y_pred
        out[NB + n] = sigmoidf_(redB[0]);              // context_pred
    }
}

// ---------------------------------------------------------------------------
extern "C" void kernel_launch(void* const* d_in, const int* in_sizes, int n_in,
                              void* d_out, int out_size, void* d_ws, size_t ws_size,
                              hipStream_t stream)
{
    const float* hidden = (const float*)d_in[0];   // [64, 2048, 768]
    const float* ctx    = (const float*)d_in[1];   // [64, 1, 768]
    const float* w_att  = (const float*)d_in[2];   // [768]
    const float* w_rec  = (const float*)d_in[3];   // [768]
    const float* b_rec  = (const float*)d_in[4];   // [1]
    float*       out    = (float*)d_out;           // 64 + 64 + 64*768
    float*       ws     = (float*)d_ws;            // needs 64*8*770*4 = 1.58 MB

    dim3 grid1(NB, CHUNKS);
    attn_flash_partial<<<grid1, BLOCK, 0, stream>>>(hidden, w_att, ws);
    attn_combine<<<NB, BLOCK, 0, stream>>>(ws, ctx, w_rec, b_rec, out);

    (void)in_sizes; (void)n_in; (void)out_size; (void)ws_size;
}